// HardMoEClassifier_67388036874691
// MI455X (gfx1250) — compile-verified
//
#include <hip/hip_runtime.h>

typedef float v2f __attribute__((ext_vector_type(2)));
typedef float v8f __attribute__((ext_vector_type(8)));

#define D_DIM          1024
#define ROWS_PER_WAVE  16
#define WAVES_PER_BLK  8
#define ROWS_PER_BLK   (ROWS_PER_WAVE * WAVES_PER_BLK)   // 128
#define SCORE_W        20   // 18 cols + 2 halves of the VALU tail, padded

__global__ __launch_bounds__(256, 2)
void HardMoEClassifier_kernel(const float* __restrict__ x,         // [B,1024]
                              const float* __restrict__ gate_w,    // [6,1024]
                              const float* __restrict__ gate_b,    // [6]
                              const float* __restrict__ expert_w,  // [6,2,1024] flat [12,1024]
                              const float* __restrict__ expert_b,  // [6,2] flat [12]
                              float* __restrict__ out,             // [B,2]
                              int nrows)
{
    __shared__ float scores[ROWS_PER_BLK][SCORE_W];

    const int tid  = threadIdx.x;
    const int wave = tid >> 5;
    const int lane = tid & 31;
    const int n    = lane & 15;   // N-col (B) / M-row (A) index within tile
    const int h    = lane >> 4;   // half selects K sub-offset {0,2}
    const int koff = h * 2;

    const int rowBase = blockIdx.x * ROWS_PER_BLK + wave * ROWS_PER_WAVE;
    int arow = rowBase + n;
    if (arow >= nrows) arow = nrows - 1;   // clamp: keep EXEC all-ones for WMMA

    // A stream: lane supplies X[arow][4k+koff .. 4k+koff+1]  (b64 per step)
    const float* pa = x + (size_t)arow * D_DIM + koff;

    // B stream, combined-weight cols 0..15: 0-5 = gate rows, 6-15 = expert flat rows 0-9
    const float* wrow = (n < 6) ? (gate_w + n * D_DIM)
                                : (expert_w + (n - 6) * D_DIM);
    const float* pb = wrow + koff;

    // VALU tail streams: combined cols 16,17 = expert flat rows 10,11
    const float* pw16 = expert_w + 10 * D_DIM + koff;
    const float* pw17 = expert_w + 11 * D_DIM + koff;

    v8f acc = {};
    float p16 = 0.0f, p17 = 0.0f;

    for (int ko = 0; ko < 32; ++ko) {           // 32 * 8 K-steps * K=4 -> D=1024
        // stay one 128B line ahead on the streaming activation
        __builtin_prefetch((const void*)(pa + ko * 32 + 32), 0, 1);
        #pragma unroll
        for (int ki = 0; ki < 8; ++ki) {
            const int k4 = (ko * 8 + ki) * 4;
            v2f a  = __builtin_nontemporal_load((const v2f*)(pa   + k4)); // read-once: NT
            v2f b  = *(const v2f*)(pb   + k4);                            // hot in L2: RT
            v2f w6 = *(const v2f*)(pw16 + k4);
            v2f w7 = *(const v2f*)(pw17 + k4);
            // D = A(16x4 f32) x B(4x16 f32) + C  -> v_wmma_f32_16x16x4_f32
            acc = __builtin_amdgcn_wmma_f32_16x16x4_f32(
                    false, a, false, b, (short)0, acc, false, false);
            p16 = fmaf(a.x, w6.x, fmaf(a.y, w6.y, p16));
            p17 = fmaf(a.x, w7.x, fmaf(a.y, w7.y, p17));
        }
    }

    // Scatter C: VGPR v holds (M = 8*h + v, N = n)
    const int rl = wave * ROWS_PER_WAVE;
    #pragma unroll
    for (int v = 0; v < 8; ++v)
        scores[rl + h * 8 + v][n] = acc[v];
    // VALU tail partials: row = n, halves h=0/1 land in cols 16/17 and 18/19
    scores[rl + n][16 + 2 * h] = p16;
    scores[rl + n][17 + 2 * h] = p17;

    __syncthreads();

    if (tid < ROWS_PER_BLK) {
        const int rg = blockIdx.x * ROWS_PER_BLK + tid;
        if (rg < nrows) {
            const float* s = scores[tid];
            // argmax over 6 gate logits (strict > keeps first max, matches jnp.argmax)
            float best = s[0] + gate_b[0];
            int bi = 0;
            #pragma unroll
            for (int e = 1; e < 6; ++e) {
                float g = s[e] + gate_b[e];
                if (g > best) { best = g; bi = e; }
            }
            const int f0 = 6 + bi * 2;
            const int f1 = f0 + 1;
            float o0 = (f0 < 16) ? s[f0] : (s[f0] + s[f0 + 2]);
            float o1 = (f1 < 16) ? s[f1] : (s[f1] + s[f1 + 2]);
            o0 += expert_b[bi * 2 + 0];
            o1 += expert_b[bi * 2 + 1];
            *(v2f*)(out + (size_t)rg * 2) = (v2f){o0, o1};
        }
    }
}

extern "C" void kernel_launch(void* const* d_in, const int* in_sizes, int n_in,
                              void* d_out, int out_size, void* d_ws, size_t ws_size,
                              hipStream_t stream) {
    const float* x        = (const float*)d_in[0];
    const float* gate_w   = (const float*)d_in[1];
    const float* gate_b   = (const float*)d_in[2];
    const float* expert_w = (const float*)d_in[3];
    const float* expert_b = (const float*)d_in[4];
    float* out = (float*)d_out;

    const int nrows = in_sizes[0] / D_DIM;
    if (nrows <= 0) return;
    const int grid = (nrows + ROWS_PER_BLK - 1) / ROWS_PER_BLK;

    HardMoEClassifier_kernel<<<grid, 256, 0, stream>>>(
        x, gate_w, gate_b, expert_w, expert_b, out, nrows);
}